// SwaVModelPairwiseMMD_44109314130141
// MI455X (gfx1250) — compile-verified
//
#include <hip/hip_runtime.h>
#include <hip/hip_bf16.h>

typedef __attribute__((ext_vector_type(2))) float v2f;
typedef __attribute__((ext_vector_type(8))) float v8f;

// ---------------------------------------------------------------------------
// Kernel 1: row-normalize feats (128 rows x 128 cols) -> xn
// ---------------------------------------------------------------------------
__global__ __launch_bounds__(128) void k_normalize(const float* __restrict__ feats,
                                                   float* __restrict__ xn) {
    __shared__ float red[128];
    const int row = blockIdx.x;
    const int t   = threadIdx.x;
    float v = feats[row * 128 + t];
    red[t] = v * v;
    __syncthreads();
    for (int s = 64; s > 0; s >>= 1) {
        if (t < s) red[t] += red[t + s];
        __syncthreads();
    }
    float n = sqrtf(red[0]);
    n = fmaxf(n, 1e-8f);
    xn[row * 128 + t] = v / n;
}

// ---------------------------------------------------------------------------
// Kernel 2: two 64x64 Gram matrices via V_WMMA_F32_16X16X4_F32 (full fp32).
// One wave (32 threads) per 16x16 output tile; 32 tiles (2 grams x 4x4).
// A-layout (16x4 f32): lane l -> M = l&15, K = (l>>4)*2 + v  (v = VGPR 0/1)
// B-layout (4x16 f32): lane l -> N = l&15, K = (l>>4)*2 + v  (B = Xn^T here,
//   so the per-lane load pattern from Xn is identical to A's).
// C/D 16x16 f32: VGPR r -> M = r + (l>>4)*8, N = l&15.
// ---------------------------------------------------------------------------
__global__ __launch_bounds__(32) void k_gram_wmma(const float* __restrict__ xn,
                                                  float* __restrict__ f) {
    const int b    = blockIdx.x;   // 0..31
    const int gram = b >> 4;       // 0 = src, 1 = trgt
    const int tile = b & 15;
    const int ti   = tile >> 2;
    const int tj   = tile & 3;
    const int l    = threadIdx.x;  // 0..31
    const int m    = l & 15;
    const int kh   = l >> 4;       // 0 or 1

    const float* base = xn + gram * 64 * 128;
    const float* arow = base + (ti * 16 + m) * 128 + kh * 2;
    const float* brow = base + (tj * 16 + m) * 128 + kh * 2;

    v8f c = {};
    #pragma unroll
    for (int k0 = 0; k0 < 128; k0 += 4) {
        v2f a  = *(const v2f*)(arow + k0);
        v2f bb = *(const v2f*)(brow + k0);
        // D = A x B + C   (emits v_wmma_f32_16x16x4_f32)
        c = __builtin_amdgcn_wmma_f32_16x16x4_f32(
                /*neg_a=*/false, a, /*neg_b=*/false, bb,
                /*c_mod=*/(short)0, c, /*reuse_a=*/false, /*reuse_b=*/false);
    }

    float* out = f + gram * 4096;      // f layout: src_d (64*64) then trgt_d
    const int col = tj * 16 + m;
    #pragma unroll
    for (int r = 0; r < 8; ++r) {
        const int row = ti * 16 + kh * 8 + r;
        out[row * 64 + col] = c[r];
    }
}

// ---------------------------------------------------------------------------
// Kernel 3: D_mean = 2*Var(f) exactly; store exp2 scale = -1/(4*Dmean*ln2)
// ---------------------------------------------------------------------------
__global__ __launch_bounds__(256) void k_stats(const float* __restrict__ f,
                                               float* __restrict__ scale) {
    __shared__ float s1[256], s2[256];
    const int t = threadIdx.x;
    float a = 0.f, b = 0.f;
    for (int j = t; j < 8192; j += 256) {
        float v = f[j];
        a += v;
        b += v * v;
    }
    s1[t] = a; s2[t] = b;
    __syncthreads();
    for (int s = 128; s > 0; s >>= 1) {
        if (t < s) { s1[t] += s1[t + s]; s2[t] += s2[t + s]; }
        __syncthreads();
    }
    if (t == 0) {
        const float inv = 1.0f / 8192.0f;
        float mean  = s1[0] * inv;
        float msq   = s2[0] * inv;
        float dmean = 2.0f * (msq - mean * mean);
        dmean = fmaxf(dmean, 1e-30f);
        // s = exp(-D/(4*Dmean)) = exp2f(D * scale)
        scale[0] = -1.0f / (4.0f * dmean * 0.69314718055994530942f);
    }
}

// ---------------------------------------------------------------------------
// Kernel 4: main pair loop. 1024 blocks x 256 threads, 8 rows per block.
// f staged in LDS (32KB). Per pair: 1 v_exp_f32 + 4 mults gives all 5 alphas:
//   K = s + s^2 + s^4 + s^8 + s^16,  s = exp(-D/(4*Dmean))
// Row-group weights applied per row; diagonal fixed in finalize.
// ---------------------------------------------------------------------------
__global__ __launch_bounds__(256) void k_pairs(const float* __restrict__ f,
                                               const float* __restrict__ scale,
                                               float* __restrict__ partials) {
    __shared__ float sF[8192];
    __shared__ float red[256];
    const int t = threadIdx.x;
    for (int j = t; j < 8192; j += 256) sF[j] = f[j];
    __syncthreads();

    const float c  = scale[0];
    const float ss =  1.0f / (4096.0f * 4095.0f);
    const float st = -1.0f / (4096.0f * 4096.0f);

    float acc = 0.0f;
    const int i0 = blockIdx.x * 8;
    for (int r = 0; r < 8; ++r) {
        const int i  = i0 + r;
        const float fi = sF[i];
        float sumS = 0.0f, sumT = 0.0f;
        #pragma unroll 4
        for (int j = t; j < 4096; j += 256) {
            float d  = fi - sF[j];
            float s  = exp2f(d * d * c);
            float p2 = s * s, p4 = p2 * p2, p8 = p4 * p4, p16 = p8 * p8;
            sumS += s + p2 + p4 + p8 + p16;
        }
        #pragma unroll 4
        for (int j = 4096 + t; j < 8192; j += 256) {
            float d  = fi - sF[j];
            float s  = exp2f(d * d * c);
            float p2 = s * s, p4 = p2 * p2, p8 = p4 * p4, p16 = p8 * p8;
            sumT += s + p2 + p4 + p8 + p16;
        }
        const bool inS = (i < 4096);
        const float wS = inS ? ss : st;
        const float wT = inS ? st : ss;
        acc += wS * sumS + wT * sumT;
    }

    red[t] = acc;
    __syncthreads();
    for (int s = 128; s > 0; s >>= 1) {
        if (t < s) red[t] += red[t + s];
        __syncthreads();
    }
    if (t == 0) partials[blockIdx.x] = red[0];
}

// ---------------------------------------------------------------------------
// Kernel 5: reduce 1024 partials, apply diagonal correction + constant term.
// loss = sum(partials) - 8192 * 5 * ss + 2/(b-1)
// ---------------------------------------------------------------------------
__global__ __launch_bounds__(256) void k_finalize(const float* __restrict__ partials,
                                                  float* __restrict__ out) {
    __shared__ float red[256];
    const int t = threadIdx.x;
    float a = 0.0f;
    for (int j = t; j < 1024; j += 256) a += partials[j];
    red[t] = a;
    __syncthreads();
    for (int s = 128; s > 0; s >>= 1) {
        if (t < s) red[t] += red[t + s];
        __syncthreads();
    }
    if (t == 0) {
        const float ss = 1.0f / (4096.0f * 4095.0f);
        out[0] = red[0] - 8192.0f * 5.0f * ss + 2.0f / 4095.0f;
    }
}

// ---------------------------------------------------------------------------
// Workspace layout (floats):
//   [0      .. 16383]  xn  (normalized feats, 128x128)
//   [16384  .. 24575]  f   (src_d 4096 then trgt_d 4096)
//   [24576]            exp2 scale
//   [24704  .. 25727]  block partials (1024)
// Total ~103 KB.
// ---------------------------------------------------------------------------
extern "C" void kernel_launch(void* const* d_in, const int* in_sizes, int n_in,
                              void* d_out, int out_size, void* d_ws, size_t ws_size,
                              hipStream_t stream) {
    const float* feats = (const float*)d_in[0];
    float* w        = (float*)d_ws;
    float* xn       = w;
    float* f        = w + 16384;
    float* scale    = w + 24576;
    float* partials = w + 24704;
    float* out      = (float*)d_out;

    k_normalize<<<128,  128, 0, stream>>>(feats, xn);
    k_gram_wmma<<<32,    32, 0, stream>>>(xn, f);
    k_stats    <<<1,    256, 0, stream>>>(f, scale);
    k_pairs    <<<1024, 256, 0, stream>>>(f, scale, partials);
    k_finalize <<<1,    256, 0, stream>>>(partials, out);
}